// MambaBlock_30855045055048
// MI455X (gfx1250) — compile-verified
//
#include <hip/hip_runtime.h>
#include <hip/hip_bf16.h>
#include <stdint.h>

// Mamba block forward for MI455X (gfx1250, wave32).
// Sizes fixed by the reference:
#define D_MODEL 768
#define D_INNER 1536
#define D_STATE 16
#define D_CONV  4
#define DT_RANK 48
#define SEQ     2048

typedef float v2f __attribute__((ext_vector_type(2)));
typedef float v8f __attribute__((ext_vector_type(8)));
typedef unsigned int uv4 __attribute__((ext_vector_type(4)));
typedef int iv4 __attribute__((ext_vector_type(4)));
typedef int iv8 __attribute__((ext_vector_type(8)));

#if __has_builtin(__builtin_amdgcn_tensor_load_to_lds) && __has_builtin(__builtin_amdgcn_s_wait_tensorcnt)
#define USE_TDM 1
#else
#define USE_TDM 0
#endif

#define KC 256           // K chunk staged in LDS
#define LDS_STRIDE 258   // 256 data dwords + 2 pad dwords per row (TDM pad_amount code 1)

__device__ __forceinline__ float silu_f(float x) { return x / (1.0f + expf(-x)); }
__device__ __forceinline__ float softplus_f(float x) {
  return (x > 20.0f) ? x : log1pf(expf(x));
}

// Generic fp32 WMMA GEMM: C[M x N] = ep( A[M x K] * W[N x K]^T ),
// ep==0: identity; ep==1: softplus(val + bias[n]).
// Block: 256 threads = 8 waves; tile 128(M) x 16(N); W panel staged in LDS.
__global__ __launch_bounds__(256) void gemm_wmma_f32(
    const float* __restrict__ A, int lda,
    const float* __restrict__ W,       // N x K row-major
    float* __restrict__ C, int ldc,
    int M, int N, int K,
    int epilogue, const float* __restrict__ bias)
{
  __shared__ float ldsB[16 * LDS_STRIDE];
  const int tid  = threadIdx.x;
  const int lane = tid & 31;
  const int wave = tid >> 5;
  const int n_base = blockIdx.x * 16;
  const int m_base = blockIdx.y * 128 + wave * 16;

  // A fragment addressing (ISA 32-bit A 16x4 layout):
  // lane m = lane&15 ; VGPR v holds K = (lane>>4)*2 + v
  const int am = m_base + (lane & 15);
  const int ak = (lane >> 4) * 2;
  const float* Arow = A + (size_t)am * lda;

  v8f acc = {};

  for (int kc = 0; kc < K; kc += KC) {
    const int chunk = (K - kc < KC) ? (K - kc) : KC;
    __syncthreads();  // LDS reuse fence

    // Prefetch next W panel rows (emits global_prefetch_b8)
    if (wave == 0 && kc + KC < K) {
      __builtin_prefetch(W + (size_t)(n_base + (lane & 15)) * K + kc + KC, 0, 0);
    }

    bool tdm_done = false;
#if USE_TDM
    if (chunk == KC) {
      if (wave == 0) {
        // Tensor DMA: 16 rows x 256 cols of W into LDS, 2-dword pad per row
        // -> LDS row stride 258 dwords (bank-conflict-free fragment reads).
        uint64_t ga = (uint64_t)(uintptr_t)(W + (size_t)n_base * K + kc);
        uv4 g0;
        g0[0] = 1u;                                   // count=1, user descriptor
        g0[1] = 0u;                                   // lds_addr: ldsB at LDS offset 0
        g0[2] = (unsigned int)(ga & 0xFFFFFFFFu);     // global_addr lo
        g0[3] = (unsigned int)((ga >> 32) & 0x01FFFFFFu) | (2u << 30); // addr hi | type=2
        iv8 g1;
        g1[0] = (2 << 16)        // data_size = 4B
              | (1 << 20)        // pad_enable
              | (7 << 22)        // pad_interval code 7 = 256 dwords
              | (1 << 25);       // pad_amount code 1 = 2 dwords
        g1[1] = (KC & 0xFFFF) << 16;   // tensor_dim0 lo16 (=256)
        g1[2] = (16 << 16);            // tensor_dim0 hi=0 | tensor_dim1 lo = 16
        g1[3] = (KC << 16);            // tensor_dim1 hi=0 | tile_dim0 = 256
        g1[4] = 16;                    // tile_dim1 = 16 rows, tile_dim2 = 0
        g1[5] = K;                     // tensor_dim0_stride = row stride (elements)
        g1[6] = 0;
        g1[7] = 0;
        iv4 gz4 = {0, 0, 0, 0};
        iv8 gz8 = {0, 0, 0, 0, 0, 0, 0, 0};
        // 6-arg toolchain form: (g0, g1, g2, g3, g4, cpol)
        __builtin_amdgcn_tensor_load_to_lds(g0, g1, gz4, gz4, gz8, 0);
        __builtin_amdgcn_s_wait_tensorcnt(0);
      }
      tdm_done = true;
    }
#endif
    if (!tdm_done) {
      for (int i = tid; i < 16 * chunk; i += 256) {
        int n = i / chunk, k = i - n * chunk;
        ldsB[n * LDS_STRIDE + k] = W[(size_t)(n_base + n) * K + kc + k];
      }
    }
    __syncthreads();

    const float* Ak = Arow + kc + ak;
    const float* Bk = &ldsB[(lane & 15) * LDS_STRIDE + ak];
    #pragma unroll 4
    for (int k = 0; k < chunk; k += 4) {
      v2f a = *reinterpret_cast<const v2f*>(Ak + k);       // global_load_b64
      v2f b = *reinterpret_cast<const v2f*>(Bk + k);       // ds_load_b64
      // D = A(16x4) * B(4x16) + C ; fp32 matrix pipe, preserves fp32 reference precision
      acc = __builtin_amdgcn_wmma_f32_16x16x4_f32(
          /*neg_a=*/false, a, /*neg_b=*/false, b,
          /*c_mod=*/(short)0, acc, /*reuse_a=*/false, /*reuse_b=*/false);
    }
  }

  // C/D layout: VGPR v -> row m_base + v + 8*(lane>>4), col n_base + (lane&15)
  const int n  = n_base + (lane & 15);
  const int m0 = m_base + 8 * (lane >> 4);
  const float bn = (epilogue == 1) ? bias[n] : 0.0f;
  #pragma unroll
  for (int v = 0; v < 8; ++v) {
    float val = acc[v];
    if (epilogue == 1) val = softplus_f(val + bn);
    C[(size_t)(m0 + v) * ldc + n] = val;
  }
}

// Depthwise causal conv (4 taps) + bias + SiLU over the first D_INNER cols of xz.
__global__ __launch_bounds__(256) void conv_silu_kernel(
    const float* __restrict__ xz, const float* __restrict__ cw,
    const float* __restrict__ cb, float* __restrict__ xba)
{
  int idx = blockIdx.x * 256 + threadIdx.x;
  if (idx >= SEQ * D_INNER) return;
  int t = idx / D_INNER;
  int d = idx - t * D_INNER;
  float acc = cb[d];
  #pragma unroll
  for (int j = 0; j < D_CONV; ++j) {
    int ts = t + j - (D_CONV - 1);
    if (ts >= 0) acc += cw[d * D_CONV + j] * xz[(size_t)ts * (2 * D_INNER) + d];
  }
  xba[idx] = silu_f(acc);
}

// Selective scan: one channel d per lane, h[16] in registers; B/C projections
// (shared by all channels) staged per 64-step chunk in LDS. Fuses the
// (y + x*D) * silu(z) epilogue so GEMM4 can consume y2 directly.
__global__ __launch_bounds__(256) void scan_kernel(
    const float* __restrict__ xdbl,   // SEQ x (DT_RANK + 2*D_STATE)
    const float* __restrict__ dtp,    // SEQ x D_INNER (post-softplus dt)
    const float* __restrict__ xba,    // SEQ x D_INNER (post-conv SiLU)
    const float* __restrict__ xz,     // SEQ x 2*D_INNER (z at +D_INNER)
    const float* __restrict__ A_log,  // D_INNER x D_STATE
    const float* __restrict__ Dp,     // D_INNER
    float* __restrict__ y2)           // SEQ x D_INNER
{
  const int d   = blockIdx.x * 256 + threadIdx.x;
  const int tid = threadIdx.x;
  float Av[D_STATE], h[D_STATE];
  #pragma unroll
  for (int n = 0; n < D_STATE; ++n) {
    Av[n] = -expf(A_log[d * D_STATE + n]);
    h[n]  = 0.0f;
  }
  const float Dd = Dp[d];
  __shared__ float lbc[64 * 32];   // per t: B[16] then C[16]

  for (int t0 = 0; t0 < SEQ; t0 += 64) {
    __syncthreads();
    for (int i = tid; i < 64 * 32; i += 256) {
      int tt = i >> 5, j = i & 31;
      lbc[i] = xdbl[(size_t)(t0 + tt) * (DT_RANK + 2 * D_STATE) + DT_RANK + j];
    }
    __syncthreads();
    if (t0 + 64 < SEQ) {
      __builtin_prefetch(&xdbl[(size_t)(t0 + 64) * (DT_RANK + 2 * D_STATE) + DT_RANK], 0, 0);
    }
    for (int tt = 0; tt < 64; ++tt) {
      const int t = t0 + tt;
      const float dtv = dtp[(size_t)t * D_INNER + d];
      const float xv  = xba[(size_t)t * D_INNER + d];
      const float zv  = xz[(size_t)t * (2 * D_INNER) + D_INNER + d];
      const float* bc = &lbc[tt * 32];
      float y = 0.0f;
      #pragma unroll
      for (int n = 0; n < D_STATE; ++n) {
        float dA = expf(dtv * Av[n]);
        h[n] = dA * h[n] + (dtv * bc[n]) * xv;
        y += h[n] * bc[16 + n];
      }
      y2[(size_t)t * D_INNER + d] = (y + xv * Dd) * silu_f(zv);
    }
  }
}

extern "C" void kernel_launch(void* const* d_in, const int* in_sizes, int n_in,
                              void* d_out, int out_size, void* d_ws, size_t ws_size,
                              hipStream_t stream) {
  (void)in_sizes; (void)n_in; (void)out_size; (void)ws_size;
  const float* x     = (const float*)d_in[0];
  const float* W_in  = (const float*)d_in[1];
  const float* cw    = (const float*)d_in[2];
  const float* cb    = (const float*)d_in[3];
  const float* W_x   = (const float*)d_in[4];
  const float* W_dt  = (const float*)d_in[5];
  const float* b_dt  = (const float*)d_in[6];
  const float* A_log = (const float*)d_in[7];
  const float* Dp    = (const float*)d_in[8];
  const float* W_out = (const float*)d_in[9];

  float* ws   = (float*)d_ws;
  float* xz   = ws;                         // SEQ*3072
  float* xba  = xz   + (size_t)SEQ * 3072;  // SEQ*1536
  float* xdbl = xba  + (size_t)SEQ * 1536;  // SEQ*80
  float* dtp  = xdbl + (size_t)SEQ * 80;    // SEQ*1536
  float* y2   = dtp  + (size_t)SEQ * 1536;  // SEQ*1536
  // total ws: ~63.6 MB of fp32 scratch

  // 1) xz = x @ W_in^T            (2048 x 3072, K=768)
  gemm_wmma_f32<<<dim3(3072 / 16, SEQ / 128), 256, 0, stream>>>(
      x, D_MODEL, W_in, xz, 2 * D_INNER, SEQ, 2 * D_INNER, D_MODEL, 0, nullptr);
  // 2) xba = silu(conv(xb) + cb)
  conv_silu_kernel<<<(SEQ * D_INNER) / 256, 256, 0, stream>>>(xz, cw, cb, xba);
  // 3) xdbl = xba @ W_x^T         (2048 x 80, K=1536)
  gemm_wmma_f32<<<dim3(80 / 16, SEQ / 128), 256, 0, stream>>>(
      xba, D_INNER, W_x, xdbl, DT_RANK + 2 * D_STATE, SEQ, DT_RANK + 2 * D_STATE,
      D_INNER, 0, nullptr);
  // 4) dtp = softplus(xdbl[:, :48] @ W_dt^T + b_dt)   (2048 x 1536, K=48)
  gemm_wmma_f32<<<dim3(D_INNER / 16, SEQ / 128), 256, 0, stream>>>(
      xdbl, DT_RANK + 2 * D_STATE, W_dt, dtp, D_INNER, SEQ, D_INNER, DT_RANK,
      1, b_dt);
  // 5) selective scan + fused (y + x*D) * silu(z)
  scan_kernel<<<D_INNER / 256, 256, 0, stream>>>(xdbl, dtp, xba, xz, A_log, Dp, y2);
  // 6) out = y2 @ W_out^T         (2048 x 768, K=1536)
  gemm_wmma_f32<<<dim3(D_MODEL / 16, SEQ / 128), 256, 0, stream>>>(
      y2, D_INNER, W_out, (float*)d_out, D_MODEL, SEQ, D_MODEL, D_INNER, 0, nullptr);
}